// SinkhornOTMatcher_45672682225772
// MI455X (gfx1250) — compile-verified
//
#include <hip/hip_runtime.h>
#include <math.h>

typedef __attribute__((ext_vector_type(2))) float v2f;
typedef __attribute__((ext_vector_type(8))) float v8f;

#define DIM 256
#define NB  128

__device__ __forceinline__ v8f wmma_f32x4(v2f a, v2f b, v8f c) {
  // D = A(16x4 f32) * B(4x16 f32) + C(16x16 f32), full f32 precision
  return __builtin_amdgcn_wmma_f32_16x16x4_f32(false, a, false, b, (short)0, c, false, false);
}

// ---------------------------------------------------------------------------
// Y[row][col] = act( X[row][:] . W[col][:] + bias[col] )   (torch Linear)
// One workgroup (8 waves) -> 64 rows x 256 cols. Wave w: rows (w&3)*16..+16,
// cols (w>>2)*128..+128 as 8 WMMA 16x16 tiles, K accumulated in steps of 4.
// ---------------------------------------------------------------------------
__global__ __launch_bounds__(256)
void proj_gemm_kernel(const float* __restrict__ X, const float* __restrict__ W,
                      const float* __restrict__ bias, float* __restrict__ Y,
                      int do_relu)
{
  const int lane = threadIdx.x & 31;
  const int wave = threadIdx.x >> 5;
  const int half = lane >> 4;     // 0: K=kb,kb+1 ; 1: K=kb+2,kb+3
  const int idx  = lane & 15;

  const int rowbase = blockIdx.x * 64 + (wave & 3) * 16;
  const int colbase = (wave >> 2) * 128;

  v8f acc[8] = {};

  const float* arow = X + (size_t)(rowbase + idx) * DIM + 2 * half;
  #pragma unroll 2
  for (int k = 0; k < DIM; k += 4) {
    v2f a = *(const v2f*)(arow + k);
    #pragma unroll
    for (int t = 0; t < 8; ++t) {
      const float* brow = W + (size_t)(colbase + t * 16 + idx) * DIM + 2 * half + k;
      acc[t] = wmma_f32x4(a, *(const v2f*)brow, acc[t]);
    }
  }

  #pragma unroll
  for (int t = 0; t < 8; ++t) {
    const int col = colbase + t * 16 + idx;
    const float bv = bias[col];
    #pragma unroll
    for (int i = 0; i < 8; ++i) {
      const int row = rowbase + half * 8 + i;   // C/D layout: VGPR i -> M=i+8*half
      float v = acc[t][i] + bv;
      if (do_relu) v = fmaxf(v, 0.0f);
      Y[(size_t)row * DIM + col] = v;
    }
  }
}

// ---------------------------------------------------------------------------
// Per-batch: C = sqrt(max(||pq||^2 + ||pr||^2 - 2 pq.pr^T, 1e-6))
//            cos = (sq.sr^T) * inv||sq|| * inv||sr||
//            10 Sinkhorn iters, then T / cost / sigmoid(-cost) / similarity.
// ---------------------------------------------------------------------------
__global__ __launch_bounds__(256)
void sinkhorn_kernel(const float* __restrict__ P,        // [16384][256] (Pq|Pr)
                     const float* __restrict__ slots_q,  // [128*64][256]
                     const float* __restrict__ slots_r,  // [128*64][256]
                     const float* __restrict__ log_eps_p,
                     float* __restrict__ out)
{
  __shared__ float Cmat[64][65];
  __shared__ float CosM[64][65];
  __shared__ float nq[64], nr[64], isq[64], isr[64];
  __shared__ float log_u[64], log_v[64];
  __shared__ float red[256];

  const int b    = blockIdx.x;
  const int t    = threadIdx.x;
  const int lane = t & 31;
  const int wave = t >> 5;
  const int half = lane >> 4;
  const int idx  = lane & 15;

  const float* Pq = P + (size_t)b * 64 * DIM;
  const float* Pr = P + (size_t)(8192 + b * 64) * DIM;
  const float* Sq = slots_q + (size_t)b * 64 * DIM;
  const float* Sr = slots_r + (size_t)b * 64 * DIM;

  // ---- phase 1: row norms (4 groups of 64 threads, one full row each) ----
  {
    const int which = t >> 6, row = t & 63;
    const float* src = (which == 0) ? (Pq + row * DIM)
                     : (which == 1) ? (Pr + row * DIM)
                     : (which == 2) ? (Sq + row * DIM)
                                    : (Sr + row * DIM);
    float s = 0.0f;
    for (int k = 0; k < DIM; ++k) { float x = src[k]; s = fmaf(x, x, s); }
    if      (which == 0) nq[row]  = s;
    else if (which == 1) nr[row]  = s;
    else if (which == 2) isq[row] = 1.0f / fmaxf(sqrtf(s), 1e-12f);
    else                 isr[row] = 1.0f / fmaxf(sqrtf(s), 1e-12f);
    if (t < 64) { log_u[t] = 0.0f; log_v[t] = 0.0f; }
  }
  __syncthreads();

  // wave tile map for 64x64: rows (w&3)*16, cols (w>>2)*32 (+0,+16)
  const int rowblk = (wave & 3) * 16;
  const int colblk = (wave >> 2) * 32;

  // ---- phase 2a: G = Pq . Pr^T  ->  C ----
  {
    v8f g0 = {}, g1 = {};
    const float* ar  = Pq + (size_t)(rowblk + idx)      * DIM + 2 * half;
    const float* b0r = Pr + (size_t)(colblk + idx)      * DIM + 2 * half;
    const float* b1r = Pr + (size_t)(colblk + 16 + idx) * DIM + 2 * half;
    #pragma unroll 2
    for (int k = 0; k < DIM; k += 4) {
      v2f a = *(const v2f*)(ar + k);
      g0 = wmma_f32x4(a, *(const v2f*)(b0r + k), g0);
      g1 = wmma_f32x4(a, *(const v2f*)(b1r + k), g1);
    }
    #pragma unroll
    for (int i = 0; i < 8; ++i) {
      const int m  = rowblk + half * 8 + i;
      const int n0 = colblk + idx, n1 = colblk + 16 + idx;
      Cmat[m][n0] = sqrtf(fmaxf(nq[m] + nr[n0] - 2.0f * g0[i], 1e-6f));
      Cmat[m][n1] = sqrtf(fmaxf(nq[m] + nr[n1] - 2.0f * g1[i], 1e-6f));
    }
  }

  // ---- phase 2b: cos = (Sq . Sr^T) scaled by inverse norms ----
  {
    v8f g0 = {}, g1 = {};
    const float* ar  = Sq + (size_t)(rowblk + idx)      * DIM + 2 * half;
    const float* b0r = Sr + (size_t)(colblk + idx)      * DIM + 2 * half;
    const float* b1r = Sr + (size_t)(colblk + 16 + idx) * DIM + 2 * half;
    #pragma unroll 2
    for (int k = 0; k < DIM; k += 4) {
      v2f a = *(const v2f*)(ar + k);
      g0 = wmma_f32x4(a, *(const v2f*)(b0r + k), g0);
      g1 = wmma_f32x4(a, *(const v2f*)(b1r + k), g1);
    }
    #pragma unroll
    for (int i = 0; i < 8; ++i) {
      const int m  = rowblk + half * 8 + i;
      const int n0 = colblk + idx, n1 = colblk + 16 + idx;
      CosM[m][n0] = g0[i] * isq[m] * isr[n0];
      CosM[m][n1] = g1[i] * isq[m] * isr[n1];
    }
  }
  __syncthreads();

  // ---- phase 3: Sinkhorn, log domain ----
  const float eps     = fminf(fmaxf(expf(log_eps_p[0]), 0.01f), 0.5f);
  const float inv_eps = 1.0f / eps;
  const float log_mu  = -logf(64.0f);
  const float log_nu  = -logf(64.0f);

  const int rr   = t >> 2;          // row/col this thread helps reduce
  const int part = t & 3;           // 4 lanes per row/col
  const int m0   = part * 16;

  for (int it = 0; it < 10; ++it) {
    // log_u[k] = log_mu - LSE_m(-C[k][m]/eps + log_v[m])
    {
      float mx = -1e30f;
      #pragma unroll
      for (int j = 0; j < 16; ++j)
        mx = fmaxf(mx, -Cmat[rr][m0 + j] * inv_eps + log_v[m0 + j]);
      mx = fmaxf(mx, __shfl_xor(mx, 1, 4));
      mx = fmaxf(mx, __shfl_xor(mx, 2, 4));
      float s = 0.0f;
      #pragma unroll
      for (int j = 0; j < 16; ++j)
        s += expf(-Cmat[rr][m0 + j] * inv_eps + log_v[m0 + j] - mx);
      s += __shfl_xor(s, 1, 4);
      s += __shfl_xor(s, 2, 4);
      if (part == 0) log_u[rr] = log_mu - (mx + logf(s));
    }
    __syncthreads();
    // log_v[m] = log_nu - LSE_k(-C[k][m]/eps + log_u[k])
    {
      float mx = -1e30f;
      #pragma unroll
      for (int j = 0; j < 16; ++j)
        mx = fmaxf(mx, -Cmat[m0 + j][rr] * inv_eps + log_u[m0 + j]);
      mx = fmaxf(mx, __shfl_xor(mx, 1, 4));
      mx = fmaxf(mx, __shfl_xor(mx, 2, 4));
      float s = 0.0f;
      #pragma unroll
      for (int j = 0; j < 16; ++j)
        s += expf(-Cmat[m0 + j][rr] * inv_eps + log_u[m0 + j] - mx);
      s += __shfl_xor(s, 1, 4);
      s += __shfl_xor(s, 2, 4);
      if (part == 0) log_v[rr] = log_nu - (mx + logf(s));
    }
    __syncthreads();
  }

  // ---- phase 4: T, cost, sigmoid(-cost), similarity ----
  float costp = 0.0f, simp = 0.0f;
  {
    const float lu = log_u[rr];
    #pragma unroll
    for (int j = 0; j < 16; ++j) {
      const int m = m0 + j;
      const float c  = Cmat[rr][m];
      const float Tv = expf(lu - c * inv_eps + log_v[m]);
      out[(size_t)b * 4096 + (size_t)rr * 64 + m] = Tv;
      costp = fmaf(Tv, c, costp);
      simp  = fmaf(Tv, CosM[rr][m], simp);
    }
  }

  red[t] = costp; __syncthreads();
  for (int off = 128; off > 0; off >>= 1) {
    if (t < off) red[t] += red[t + off];
    __syncthreads();
  }
  const float cost_b = red[0];
  __syncthreads();
  red[t] = simp; __syncthreads();
  for (int off = 128; off > 0; off >>= 1) {
    if (t < off) red[t] += red[t + off];
    __syncthreads();
  }
  if (t == 0) {
    const size_t TOFF = (size_t)NB * 4096;   // 524288
    out[TOFF + b]           = cost_b;
    out[TOFF + NB + b]      = 1.0f / (1.0f + expf(cost_b));   // sigmoid(-cost)
    out[TOFF + 2 * NB + b]  = red[0];                         // similarity
  }
}

// ---------------------------------------------------------------------------
extern "C" void kernel_launch(void* const* d_in, const int* in_sizes, int n_in,
                              void* d_out, int out_size, void* d_ws, size_t ws_size,
                              hipStream_t stream) {
  const float* slots_q = (const float*)d_in[0];   // [128,64,256]
  const float* slots_r = (const float*)d_in[1];   // [128,64,256]
  const float* W1      = (const float*)d_in[2];   // [256,256]
  const float* b1      = (const float*)d_in[3];   // [256]
  const float* W2      = (const float*)d_in[4];   // [256,256]
  const float* b2      = (const float*)d_in[5];   // [256]
  const float* log_eps = (const float*)d_in[6];   // [1]
  float* out = (float*)d_out;

  float* H = (float*)d_ws;                        // [16384][256] hidden
  float* P = H + (size_t)16384 * DIM;             // [16384][256] projected (Pq|Pr)

  // layer 1: H = relu(X @ W1^T + b1)
  proj_gemm_kernel<<<128, 256, 0, stream>>>(slots_q, W1, b1, H, 1);
  proj_gemm_kernel<<<128, 256, 0, stream>>>(slots_r, W1, b1, H + (size_t)8192 * DIM, 1);
  // layer 2: P = H @ W2^T + b2   (both halves in one launch)
  proj_gemm_kernel<<<256, 256, 0, stream>>>(H, W2, b2, P, 0);
  // per-batch cost matrix + Sinkhorn + outputs
  sinkhorn_kernel<<<128, 256, 0, stream>>>(P, slots_q, slots_r, log_eps, out);
}